// RandomWalk_43757126811920
// MI455X (gfx1250) — compile-verified
//
#include <hip/hip_runtime.h>

typedef __attribute__((ext_vector_type(2))) float v2f;
typedef __attribute__((ext_vector_type(8))) float v8f;

__global__ void fill_zero_f32(float* __restrict__ p, long n) {
  long i = (long)blockIdx.x * blockDim.x + threadIdx.x;
  if (i < n) p[i] = 0.0f;
}

__global__ void fill_zero_f32v4(float4* __restrict__ p, long n4) {
  long i = (long)blockIdx.x * blockDim.x + threadIdx.x;
  if (i < n4) p[i] = make_float4(0.f, 0.f, 0.f, 0.f);
}

// deg[row[e]] += attr[e]  (segment_sum over source nodes)
__global__ void degree_kernel(const int* __restrict__ row,
                              const float* __restrict__ attr,
                              float* __restrict__ deg, int E) {
  int e = blockIdx.x * blockDim.x + threadIdx.x;
  if (e < E) unsafeAtomicAdd(&deg[row[e]], attr[e]);
}

// norm[e] = attr[e] / max(deg[row[e]], 1e-12)
__global__ void norm_kernel(const int* __restrict__ row,
                            const float* __restrict__ attr,
                            const float* __restrict__ deg,
                            float* __restrict__ norm, int E) {
  int e = blockIdx.x * blockDim.x + threadIdx.x;
  if (e < E) norm[e] = attr[e] / fmaxf(deg[row[e]], 1e-12f);
}

__global__ void onehot_kernel(const int* __restrict__ target,
                              float* __restrict__ h, int N, int C) {
  int n = blockIdx.x * blockDim.x + threadIdx.x;
  if (n < N) {
    int t = target[n];
    if (t >= 0 && t < C) h[(long)n * C + t] = 1.0f;
  }
}

// 16 threads per edge; lane cl covers classes {cl, cl+16, cl+32}.
// C=48 specialization: three straight-line global_atomic_add_f32 per lane.
__global__ void spmm_scatter48(const int* __restrict__ row,
                               const int* __restrict__ col,
                               const float* __restrict__ norm,
                               const float* __restrict__ h_cur,
                               float* __restrict__ h_next, int E) {
  long gid = (long)blockIdx.x * blockDim.x + threadIdx.x;
  int cl = (int)(gid & 15);
  long e = gid >> 4;
  if (e >= E) return;
  int r = row[e];
  int c = col[e];
  float w = norm[e];
  const float* hs = h_cur + (long)r * 48 + cl;
  float*       hd = h_next + (long)c * 48 + cl;
  unsafeAtomicAdd(hd +  0, w * hs[ 0]);
  unsafeAtomicAdd(hd + 16, w * hs[16]);
  unsafeAtomicAdd(hd + 32, w * hs[32]);
}

// Generic-C fallback (not used for the reference shape).
__global__ void spmm_scatter(const int* __restrict__ row,
                             const int* __restrict__ col,
                             const float* __restrict__ norm,
                             const float* __restrict__ h_cur,
                             float* __restrict__ h_next, int E, int C) {
  long gid = (long)blockIdx.x * blockDim.x + threadIdx.x;
  int cl = (int)(gid & 15);
  long e = gid >> 4;
  if (e >= E) return;
  int r = row[e];
  int c = col[e];
  float w = norm[e];
  const float* hs = h_cur + (long)r * C;
  float*       hd = h_next + (long)c * C;
  for (int off = cl; off < C; off += 16)
    unsafeAtomicAdd(hd + off, w * hs[off]);
}

// out_tile(16x16) += H_tile(16x16) * diag(w[c,s]) via 4x V_WMMA_F32_16X16X4_F32.
// One wave per (node-tile, class-tile). FP32-native WMMA -> no precision loss.
// A layout (16x4 f32): lanes 0-15 hold K=0,1; lanes 16-31 hold K=2,3.
// B layout (4x16 f32): mirrored; C/D: 8 VGPRs, lane = column, vgpr r -> row r(+8).
// Full-tile fast path keeps EXEC all-1s with straight-line loads/stores.
__global__ void combine_wmma(const float* __restrict__ h,
                             const float* __restrict__ w,   // [C, S] row-major
                             float* __restrict__ out,
                             int s, int S, int N, int C,
                             int tiles_n, int tiles_c) {
  int wave = (int)(((long)blockIdx.x * blockDim.x + threadIdx.x) >> 5);
  int lane = threadIdx.x & 31;
  if (wave >= tiles_n * tiles_c) return;   // whole-wave uniform: EXEC stays all-1s
  int nt = wave / tiles_c, ct = wave % tiles_c;
  int n0 = nt * 16, c0 = ct * 16;
  int half = lane >> 4, li = lane & 15;
  bool full = (n0 + 16 <= N);              // wave-uniform

  float ws = w[(c0 + li) * S + s];

  int arow = n0 + li;
  if (arow >= N) arow = N - 1;             // clamp: loads stay EXEC-all-1
  const float* hp = h + (long)arow * C + c0;
  float* op = out + ((long)n0 + 8 * half) * C + c0 + li;

  v8f acc;
  if (full) {
#pragma unroll
    for (int r = 0; r < 8; ++r) acc[r] = op[(long)r * C];
  } else {
#pragma unroll
    for (int r = 0; r < 8; ++r) {
      int m = n0 + r + 8 * half; if (m >= N) m = N - 1;
      acc[r] = out[(long)m * C + c0 + li];
    }
  }

#pragma unroll
  for (int kk = 0; kk < 4; ++kk) {
    int k0 = 4 * kk + 2 * half;
    v2f a = *(const v2f*)(hp + k0);        // 8B-aligned: global_load_b64
    v2f b;
    b.x = (li == k0)     ? ws : 0.0f;      // diagonal weight sub-block
    b.y = (li == k0 + 1) ? ws : 0.0f;
    acc = __builtin_amdgcn_wmma_f32_16x16x4_f32(false, a, false, b,
                                                (short)0, acc, false, false);
  }

  if (full) {
#pragma unroll
    for (int r = 0; r < 8; ++r) op[(long)r * C] = acc[r];
  } else {
#pragma unroll
    for (int r = 0; r < 8; ++r) {
      int m = n0 + r + 8 * half;
      if (m < N) out[(long)m * C + c0 + li] = acc[r];
    }
  }
}

extern "C" void kernel_launch(void* const* d_in, const int* in_sizes, int n_in,
                              void* d_out, int out_size, void* d_ws, size_t ws_size,
                              hipStream_t stream) {
  const int*   edge_index = (const int*)d_in[0];   // [2, E]
  const float* edge_attr  = (const float*)d_in[1]; // [E]
  const int*   target     = (const int*)d_in[2];   // [N]
  const float* weight     = (const float*)d_in[3]; // [C, S]

  const int E = in_sizes[1];
  const int N = in_sizes[2];
  const int S = 10;
  const int C = in_sizes[3] / S;          // 48
  const int tiles_n = (N + 15) / 16;
  const int tiles_c = C / 16;

  const int* row = edge_index;
  const int* col = edge_index + E;

  float* deg  = (float*)d_ws;             // N
  float* norm = deg + N;                  // E
  float* h_a  = norm + E;                 // N*C
  float* h_b  = h_a + (size_t)N * C;      // N*C
  float* out  = (float*)d_out;

  const int BT = 256;
  auto blocks = [&](long n) { return (int)((n + BT - 1) / BT); };
  auto zero = [&](float* p, long n) {
    if ((n & 3) == 0)
      fill_zero_f32v4<<<blocks(n >> 2), BT, 0, stream>>>((float4*)p, n >> 2);
    else
      fill_zero_f32<<<blocks(n), BT, 0, stream>>>(p, n);
  };

  zero(deg, N);
  zero(h_a, (long)N * C);
  zero(out, (long)N * C);

  degree_kernel<<<blocks(E), BT, 0, stream>>>(row, edge_attr, deg, E);
  norm_kernel<<<blocks(E), BT, 0, stream>>>(row, edge_attr, deg, norm, E);
  onehot_kernel<<<blocks(N), BT, 0, stream>>>(target, h_a, N, C);

  float* hc = h_a;
  float* hn = h_b;
  long spmm_threads = (long)E * 16;
  long comb_threads = (long)tiles_n * tiles_c * 32;

  for (int s = 0; s < S; ++s) {
    zero(hn, (long)N * C);
    if (C == 48)
      spmm_scatter48<<<blocks(spmm_threads), BT, 0, stream>>>(row, col, norm, hc, hn, E);
    else
      spmm_scatter<<<blocks(spmm_threads), BT, 0, stream>>>(row, col, norm, hc, hn, E, C);
    combine_wmma<<<blocks(comb_threads), BT, 0, stream>>>(hn, weight, out,
                                                          s, S, N, C, tiles_n, tiles_c);
    float* t = hc; hc = hn; hn = t;
  }
}